// Model_21105469293024
// MI455X (gfx1250) — compile-verified
//
#include <hip/hip_runtime.h>
#include <hip/hip_bf16.h>

typedef float v2f __attribute__((ext_vector_type(2)));
typedef float v8f __attribute__((ext_vector_type(8)));
typedef int   v4i __attribute__((ext_vector_type(4)));

#define BM 128
#define BN 64
#define KC 16
#define LDA 18   // padded LDS row stride (conflict-free, keeps float-pairs 8B aligned)

// ---- CDNA5 async global->LDS staging (guarded; falls back to register path) ----
#if defined(__has_builtin)
#if __has_builtin(__builtin_amdgcn_global_load_async_to_lds_b128) && \
    __has_builtin(__builtin_amdgcn_global_load_async_to_lds_b32)
#define HAVE_ASYNC_LDS 1
#endif
#endif
#ifndef HAVE_ASYNC_LDS
#define HAVE_ASYNC_LDS 0
#endif

#if HAVE_ASYNC_LDS
// exact param types per hipcc diagnostic: src is AS1 (global), dst is AS3 (LDS)
typedef __attribute__((address_space(1))) v4i g_v4i;
typedef __attribute__((address_space(3))) v4i l_v4i;
typedef __attribute__((address_space(1))) int g_i32;
typedef __attribute__((address_space(3))) int l_i32;

static __device__ __forceinline__ void async_wait_all() {
#if __has_builtin(__builtin_amdgcn_s_wait_asynccnt)
    __builtin_amdgcn_s_wait_asynccnt(0);
#else
    asm volatile("s_wait_asynccnt 0x0" ::: "memory");
#endif
}
#endif

// ---------------------------------------------------------------------------
// Count edges per (relation, dst) — graph-only, computed once for all layers.
// ---------------------------------------------------------------------------
__global__ void count_edges_k(const int* __restrict__ dst, const int* __restrict__ ety,
                              int E, int Nn, int* __restrict__ cnt) {
    int e = blockIdx.x * blockDim.x + threadIdx.x;
    if (e < E) atomicAdd(&cnt[ety[e] * Nn + dst[e]], 1);
}

// ---------------------------------------------------------------------------
// Exclusive scan over cnt[R*N] -> offs (3-kernel: per-block scan, aux scan, add)
// ---------------------------------------------------------------------------
__global__ void scan_block_k(const int* __restrict__ cnt, int* __restrict__ offs,
                             int* __restrict__ aux, int n) {
    __shared__ int tmp[256];
    const int i = blockIdx.x * 256 + threadIdx.x;
    const int v = (i < n) ? cnt[i] : 0;
    tmp[threadIdx.x] = v;
    __syncthreads();
    for (int d = 1; d < 256; d <<= 1) {
        int t = (threadIdx.x >= d) ? tmp[threadIdx.x - d] : 0;
        __syncthreads();
        tmp[threadIdx.x] += t;
        __syncthreads();
    }
    if (i < n) offs[i] = tmp[threadIdx.x] - v;   // exclusive
    if (threadIdx.x == 255) aux[blockIdx.x] = tmp[255];
}

__global__ void scan_aux_k(int* __restrict__ aux, int nb) {
    if (blockIdx.x == 0 && threadIdx.x == 0) {
        int run = 0;
        for (int b = 0; b < nb; b++) { int t = aux[b]; aux[b] = run; run += t; }
    }
}

__global__ void add_aux_k(int* __restrict__ offs, const int* __restrict__ aux, int n) {
    int i = blockIdx.x * 256 + threadIdx.x;
    if (i < n) offs[i] += aux[blockIdx.x];
}

// ---------------------------------------------------------------------------
// Bucket-fill CSR edge list: elist[offs[seg] + pos] = src  (graph-only, once)
// ---------------------------------------------------------------------------
__global__ void fill_k(const int* __restrict__ src, const int* __restrict__ dst,
                       const int* __restrict__ ety, int E, int Nn,
                       const int* __restrict__ offs, int* __restrict__ cursor,
                       int* __restrict__ elist) {
    int e = blockIdx.x * blockDim.x + threadIdx.x;
    if (e < E) {
        int seg = ety[e] * Nn + dst[e];
        int p = atomicAdd(&cursor[seg], 1);
        elist[offs[seg] + p] = src[e];
    }
}

// ---------------------------------------------------------------------------
// Gather-based mean aggregation: one wave per dst row, no atomics.
// A[dst] = (1/max(cnt,1)) * sum_{e in seg(r,dst)} x[src[e]]
// ---------------------------------------------------------------------------
template <int DIN>
__global__ void aggregate_k(const float* __restrict__ x, const int* __restrict__ elist,
                            const int* __restrict__ offs, const int* __restrict__ cnt,
                            int Nn, int r, float* __restrict__ A) {
    const int gtid = blockIdx.x * blockDim.x + threadIdx.x;
    const int wid  = gtid >> 5;
    const int lane = gtid & 31;
    const int nw   = (gridDim.x * blockDim.x) >> 5;
    for (int d = wid; d < Nn; d += nw) {
        const int seg = r * Nn + d;
        const int o = offs[seg];
        const int c = cnt[seg];
        float4 a0 = make_float4(0.f, 0.f, 0.f, 0.f);
        float4 a1 = make_float4(0.f, 0.f, 0.f, 0.f);
        for (int j = 0; j < c; j++) {
            const float* xs = x + (size_t)elist[o + j] * DIN;
            float4 v0 = *(const float4*)(xs + lane * 4);
            a0.x += v0.x; a0.y += v0.y; a0.z += v0.z; a0.w += v0.w;
            if (DIN == 256) {
                float4 v1 = *(const float4*)(xs + 128 + lane * 4);
                a1.x += v1.x; a1.y += v1.y; a1.z += v1.z; a1.w += v1.w;
            }
        }
        const float s = 1.0f / (float)(c > 1 ? c : 1);
        float* ad = A + (size_t)d * DIN;
        a0.x *= s; a0.y *= s; a0.z *= s; a0.w *= s;
        *(float4*)(ad + lane * 4) = a0;
        if (DIN == 256) {
            a1.x *= s; a1.y *= s; a1.z *= s; a1.w *= s;
            *(float4*)(ad + 128 + lane * 4) = a1;
        }
    }
}

// ---------------------------------------------------------------------------
// fp32 WMMA GEMM: C = A*B (+bias | +C_old), optional ReLU.
// A: [M,K] row-major, B: [K,Nn] row-major, C: [M,Nn] row-major.
// Block 256 thr = 8 waves (4x2), tile 128x64, wave tile 32x32 (2x2 of 16x16).
// Full-tile blocks stage A/B via async global->LDS (ASYNCcnt); partial M-tile
// keeps the guarded register path (needs zero-fill predication).
// ---------------------------------------------------------------------------
__global__ __launch_bounds__(256) void gemm_wmma_f32(
    const float* __restrict__ A, const float* __restrict__ B, float* __restrict__ C,
    int M, int K, int Nn, const float* __restrict__ bias, int initC, int relu)
{
    __shared__ float As[BM][LDA];
    __shared__ float Bs[BN][LDA];   // B stored transposed: Bs[n][k]

    const int tid  = threadIdx.x;
    const int lane = tid & 31;
    const int w    = tid >> 5;
    const int wm   = w & 3;        // 4 waves along M
    const int wn   = w >> 2;       // 2 waves along N
    const int h    = lane >> 4;    // half-wave 0/1
    const int l16  = lane & 15;

    const int m0 = blockIdx.x * BM;
    const int n0 = blockIdx.y * BN;
    const bool full = (m0 + BM) <= M;   // block-uniform

    v8f acc[2][2];
#pragma unroll
    for (int tm = 0; tm < 2; tm++)
#pragma unroll
        for (int tn = 0; tn < 2; tn++) {
            const int col   = n0 + wn * 32 + tn * 16 + l16;
            const int rbase = m0 + wm * 32 + tm * 16 + 8 * h;
            if (initC) {
                const float bv = bias ? bias[col] : 0.0f;
#pragma unroll
                for (int v = 0; v < 8; v++) acc[tm][tn][v] = bv;
            } else if (full) {
#pragma unroll
                for (int v = 0; v < 8; v++)
                    acc[tm][tn][v] = C[(size_t)(rbase + v) * Nn + col];
            } else {
#pragma unroll
                for (int v = 0; v < 8; v++) {
                    const int row = rbase + v;
                    acc[tm][tn][v] = (row < M) ? C[(size_t)row * Nn + col] : 0.0f;
                }
            }
        }

    for (int kc = 0; kc < K; kc += KC) {
        // --- A tile (128 x 16) ---
        if (full) {
#if HAVE_ASYNC_LDS
#pragma unroll
            for (int i = tid; i < BM * (KC / 4); i += 256) {
                const int row = i >> 2;
                const int k4  = (i & 3) << 2;
                __builtin_amdgcn_global_load_async_to_lds_b128(
                    (g_v4i*)(A + (size_t)(m0 + row) * K + kc + k4),
                    (l_v4i*)(&As[row][k4]), 0, 0);
            }
#else
#pragma unroll
            for (int i = tid; i < BM * (KC / 4); i += 256) {
                const int row = i >> 2;
                const int k4  = (i & 3) << 2;
                float4 v = *(const float4*)(A + (size_t)(m0 + row) * K + kc + k4);
                As[row][k4 + 0] = v.x;
                As[row][k4 + 1] = v.y;
                As[row][k4 + 2] = v.z;
                As[row][k4 + 3] = v.w;
            }
#endif
        } else {
#pragma unroll
            for (int i = tid; i < BM * (KC / 4); i += 256) {
                const int row = i >> 2;
                const int k4  = (i & 3) << 2;
                const int grow = m0 + row;
                float4 v = make_float4(0.f, 0.f, 0.f, 0.f);
                if (grow < M) v = *(const float4*)(A + (size_t)grow * K + kc + k4);
                As[row][k4 + 0] = v.x;
                As[row][k4 + 1] = v.y;
                As[row][k4 + 2] = v.z;
                As[row][k4 + 3] = v.w;
            }
        }
        // --- B tile transposed into LDS: Bs[n][k] (always full in N) ---
#if HAVE_ASYNC_LDS
#pragma unroll
        for (int i = tid; i < BN * KC; i += 256) {
            const int n = i & (BN - 1);
            const int k = i >> 6;
            __builtin_amdgcn_global_load_async_to_lds_b32(
                (g_i32*)(B + (size_t)(kc + k) * Nn + n0 + n),
                (l_i32*)(&Bs[n][k]), 0, 0);
        }
#else
#pragma unroll
        for (int i = tid; i < BN * KC; i += 256) {
            const int n = i & (BN - 1);
            const int k = i >> 6;
            Bs[n][k] = B[(size_t)(kc + k) * Nn + n0 + n];
        }
#endif
#if HAVE_ASYNC_LDS
        async_wait_all();
#endif
        __syncthreads();

#pragma unroll
        for (int k0 = 0; k0 < KC; k0 += 4) {
            const int kk = k0 + 2 * h;
            v2f a[2], b[2];
#pragma unroll
            for (int tm = 0; tm < 2; tm++) {
                const float* p = &As[wm * 32 + tm * 16 + l16][kk];
                a[tm][0] = p[0];
                a[tm][1] = p[1];
            }
#pragma unroll
            for (int tn = 0; tn < 2; tn++) {
                const float* p = &Bs[wn * 32 + tn * 16 + l16][kk];
                b[tn][0] = p[0];
                b[tn][1] = p[1];
            }
#pragma unroll
            for (int tm = 0; tm < 2; tm++)
#pragma unroll
                for (int tn = 0; tn < 2; tn++)
                    acc[tm][tn] = __builtin_amdgcn_wmma_f32_16x16x4_f32(
                        false, a[tm], false, b[tn], (short)0, acc[tm][tn], false, false);
        }
        __syncthreads();
    }

    // --- epilogue ---
#pragma unroll
    for (int tm = 0; tm < 2; tm++)
#pragma unroll
        for (int tn = 0; tn < 2; tn++) {
            const int col   = n0 + wn * 32 + tn * 16 + l16;
            const int rbase = m0 + wm * 32 + tm * 16 + 8 * h;
            if (full) {
#pragma unroll
                for (int v = 0; v < 8; v++) {
                    float val = acc[tm][tn][v];
                    if (relu && val < 0.f) val = 0.f;
                    C[(size_t)(rbase + v) * Nn + col] = val;
                }
            } else {
#pragma unroll
                for (int v = 0; v < 8; v++) {
                    const int row = rbase + v;
                    if (row < M) {
                        float val = acc[tm][tn][v];
                        if (relu && val < 0.f) val = 0.f;
                        C[(size_t)row * Nn + col] = val;
                    }
                }
            }
        }
}

// ---------------------------------------------------------------------------
// Column mean over nodes: out[256] += sum(rows)/N
// ---------------------------------------------------------------------------
__global__ void colmean_k(const float* __restrict__ u, float* __restrict__ out,
                          int NN, int rowsPerBlk) {
    const int col  = threadIdx.x;          // 256 columns
    const int row0 = blockIdx.x * rowsPerBlk;
    float s = 0.f;
    for (int r = 0; r < rowsPerBlk; r++) {
        const int row = row0 + r;
        if (row < NN) s += u[(size_t)row * 256 + col];
    }
    atomicAdd(&out[col], s / (float)NN);
}

// ---------------------------------------------------------------------------
extern "C" void kernel_launch(void* const* d_in, const int* in_sizes, int n_in,
                              void* d_out, int out_size, void* d_ws, size_t ws_size,
                              hipStream_t stream) {
    const int IN_DIM = 128, HID = 256, R = 8;
    const int NN = in_sizes[0] / IN_DIM;     // 30000
    const int E  = in_sizes[2];              // 480000
    const int RN = R * NN;

    const float* x    = (const float*)d_in[0];
    const int*   ei   = (const int*)d_in[1];
    const int*   src  = ei;
    const int*   dst  = ei + E;
    const int*   ety  = (const int*)d_in[2];
    const float* W1   = (const float*)d_in[3];
    const float* rt1  = (const float*)d_in[4];
    const float* b1   = (const float*)d_in[5];
    const float* W2   = (const float*)d_in[6];
    const float* rt2  = (const float*)d_in[7];
    const float* b2   = (const float*)d_in[8];
    const float* W3   = (const float*)d_in[9];
    const float* rt3  = (const float*)d_in[10];
    const float* b3   = (const float*)d_in[11];
    float* out = (float*)d_out;

    // workspace layout
    float* uA     = (float*)d_ws;                    // NN*256
    float* uB     = uA + (size_t)NN * 256;           // NN*256
    float* Abuf   = uB + (size_t)NN * 256;           // NN*256 (aggregation buffer)
    int*   cnt    = (int*)(Abuf + (size_t)NN * 256); // R*NN
    int*   offs   = cnt + RN;                        // R*NN
    int*   cursor = offs + RN;                       // R*NN
    int*   elist  = cursor + RN;                     // E
    int*   aux    = elist + E;                       // scan aux (~1k)

    const int scanBlocks = (RN + 255) / 256;

    // --- CSR build: counts -> offsets -> bucket fill (graph-only, once) ---
    (void)hipMemsetAsync(cnt, 0, sizeof(int) * (size_t)RN, stream);
    count_edges_k<<<(E + 255) / 256, 256, 0, stream>>>(dst, ety, E, NN, cnt);
    scan_block_k<<<scanBlocks, 256, 0, stream>>>(cnt, offs, aux, RN);
    scan_aux_k<<<1, 64, 0, stream>>>(aux, scanBlocks);
    add_aux_k<<<scanBlocks, 256, 0, stream>>>(offs, aux, RN);
    (void)hipMemsetAsync(cursor, 0, sizeof(int) * (size_t)RN, stream);
    fill_k<<<(E + 255) / 256, 256, 0, stream>>>(src, dst, ety, E, NN, offs, cursor, elist);

    dim3 gemmGrid((NN + BM - 1) / BM, HID / BN);

    // --- one RGCN layer: out = relu?( sum_r mean_r(xin) W_r + xin*root + b ) ---
    auto layer = [&](const float* xin, int Din, const float* W, const float* root,
                     const float* bias, float* outp, int do_relu) {
        // init: out = xin @ root + bias
        gemm_wmma_f32<<<gemmGrid, 256, 0, stream>>>(
            xin, root, outp, NN, Din, HID, bias, 1, 0);
        for (int r = 0; r < R; r++) {
            if (Din == 128)
                aggregate_k<128><<<512, 256, 0, stream>>>(xin, elist, offs, cnt, NN, r, Abuf);
            else
                aggregate_k<256><<<512, 256, 0, stream>>>(xin, elist, offs, cnt, NN, r, Abuf);
            const int last = (r == R - 1);
            gemm_wmma_f32<<<gemmGrid, 256, 0, stream>>>(
                Abuf, W + (size_t)r * Din * HID, outp, NN, Din, HID,
                nullptr, 0, last && do_relu);
        }
    };

    layer(x,  IN_DIM, W1, rt1, b1, uA, 1);   // layer 1: 128 -> 256, ReLU
    layer(uA, HID,    W2, rt2, b2, uB, 1);   // layer 2: 256 -> 256, ReLU
    layer(uB, HID,    W3, rt3, b3, uA, 0);   // layer 3: 256 -> 256

    // --- mean over nodes -> [1, 256] ---
    (void)hipMemsetAsync(out, 0, sizeof(float) * 256, stream);
    const int blocks = 120;
    const int rowsPerBlk = (NN + blocks - 1) / blocks;
    colmean_k<<<blocks, 256, 0, stream>>>(uA, out, NN, rowsPerBlk);
}